// FCNNRhoValuationFunction_39977555591639
// MI455X (gfx1250) — compile-verified
//
#include <hip/hip_runtime.h>
#include <stdint.h>

#define TILE   256
#define DIMS   11
#define GRADES 10

// 16-byte vector type matching the builtin's expected pointee type.
typedef int v4i __attribute__((vector_size(16)));
typedef __attribute__((address_space(1))) v4i g_v4i;   // global
typedef __attribute__((address_space(3))) v4i l_v4i;   // LDS

// One 16-byte async copy global -> LDS (tracked by ASYNCcnt).
__device__ __forceinline__ void async_copy16(const void* gsrc, void* ldst) {
#if defined(__gfx1250__) && __has_builtin(__builtin_amdgcn_global_load_async_to_lds_b128)
  __builtin_amdgcn_global_load_async_to_lds_b128(
      (g_v4i*)const_cast<void*>(gsrc), (l_v4i*)ldst, /*offset=*/0, /*cpol=*/0);
#else
  unsigned loff = (unsigned)(unsigned long long)(l_v4i*)ldst;
  asm volatile("global_load_async_to_lds_b128 %0, %1, off"
               :: "v"(loff), "v"(gsrc) : "memory");
#endif
}

__device__ __forceinline__ void wait_async0() {
#if defined(__gfx1250__) && __has_builtin(__builtin_amdgcn_s_wait_asynccnt)
  __builtin_amdgcn_s_wait_asynccnt(0);
#else
  asm volatile("s_wait_asynccnt 0" ::: "memory");
#endif
}

__device__ __forceinline__ int rho_bin(float x1, float y1, float x2, float y2) {
  const float dx = x2 - x1;
  const float dy = y1 - y2;            // sign irrelevant once squared
  const float rho = __builtin_sqrtf(dx * dx + dy * dy);
  int id = 0;
#pragma unroll
  for (int k = 1; k <= 9; ++k) id += (rho >= (float)k * 0.1f) ? 1 : 0;
  return id;                            // 0..9
}

__global__ __launch_bounds__(TILE)
void FCNNRhoValuation_kernel(const float* __restrict__ z1,
                             const float* __restrict__ z2,
                             const float* __restrict__ dg,
                             float* __restrict__ out, int B) {
  __shared__ __align__(16) float s_z1[TILE * DIMS];    // 11264 B
  __shared__ __align__(16) float s_z2[TILE * DIMS];    // 11264 B
  __shared__ __align__(16) float s_dg[TILE * GRADES];  // 10240 B

  const int t = threadIdx.x;
  const long long row0 = (long long)blockIdx.x * TILE;

  if (row0 + TILE <= (long long)B) {
    // ---- fast path: full tile, async-staged through LDS ----
    const char* gz1 = (const char*)(z1 + row0 * DIMS);
    const char* gz2 = (const char*)(z2 + row0 * DIMS);
    const char* gdg = (const char*)(dg + row0 * GRADES);
    constexpr int ZC = (TILE * DIMS * 4) / 16;    // 704 chunks of 16 B
    constexpr int GC = (TILE * GRADES * 4) / 16;  // 640 chunks of 16 B

    for (int c = t; c < ZC; c += TILE) {
      async_copy16(gz1 + (size_t)c * 16, (char*)s_z1 + (size_t)c * 16);
      async_copy16(gz2 + (size_t)c * 16, (char*)s_z2 + (size_t)c * 16);
    }
    for (int c = t; c < GC; c += TILE) {
      async_copy16(gdg + (size_t)c * 16, (char*)s_dg + (size_t)c * 16);
    }
    wait_async0();
    __syncthreads();

    const int id = rho_bin(s_z1[t * DIMS + 0], s_z1[t * DIMS + 2],
                           s_z2[t * DIMS + 0], s_z2[t * DIMS + 2]);
    // streaming output: written once, never re-read -> non-temporal
    __builtin_nontemporal_store(s_dg[t * GRADES + id], out + row0 + t);
  } else {
    // ---- tail path: plain per-row loads ----
    const long long row = row0 + t;
    if (row < (long long)B) {
      const int id = rho_bin(z1[row * DIMS + 0], z1[row * DIMS + 2],
                             z2[row * DIMS + 0], z2[row * DIMS + 2]);
      out[row] = dg[row * GRADES + id];
    }
  }
}

extern "C" void kernel_launch(void* const* d_in, const int* in_sizes, int n_in,
                              void* d_out, int out_size, void* d_ws, size_t ws_size,
                              hipStream_t stream) {
  const float* z1 = (const float*)d_in[0];       // [B, 11]
  const float* z2 = (const float*)d_in[1];       // [B, 11]
  const float* dg = (const float*)d_in[2];       // [B, 10]
  float* out = (float*)d_out;                    // [B]
  const int B = in_sizes[0] / DIMS;              // 4,000,000
  const int nblk = (B + TILE - 1) / TILE;        // 15625
  FCNNRhoValuation_kernel<<<nblk, TILE, 0, stream>>>(z1, z2, dg, out, B);
}